// AttentionBlock_89034672046380
// MI455X (gfx1250) — compile-verified
//
#include <hip/hip_runtime.h>

// ----------------------------------------------------------------------------
// MI455X (gfx1250) fused linear + LeakyReLU + scatter-softmax.
//
// Roofline: x is 1.64 GB read once; 23.3 TB/s HBM => ~70us floor. Everything
// else (<30 MB) lives in the 192 MB L2. So: one streaming pass over x doing
// the GEMV via V_WMMA_F32_16X16X4_F32 (16 rows/wave, 32 WMMA accumulations),
// then three cheap L2-resident passes for the segment softmax.
// ----------------------------------------------------------------------------

typedef __attribute__((ext_vector_type(2))) float v2f;
typedef __attribute__((ext_vector_type(8))) float v8f;

#define F_DIM 128
#define NSEG  100000        // reference constant N (not derivable from inputs)
#define NEG_SLOPE 0.2f

// Order-preserving float<->uint mapping so segment-max can use atomicMax(u32).
__device__ __forceinline__ unsigned enc_f32(float f) {
  unsigned u = __float_as_uint(f);
  return (u & 0x80000000u) ? ~u : (u | 0x80000000u);
}
__device__ __forceinline__ float dec_f32(unsigned u) {
  return (u & 0x80000000u) ? __uint_as_float(u & 0x7FFFFFFFu)
                           : __uint_as_float(~u);
}
#define ENC_NEG_INF 0x007FFFFFu   // enc(-inf)

// ---------------------------------------------------------------------------
// Pass 0: init segment max (enc(-inf)) and segment sum (0).
// ---------------------------------------------------------------------------
__global__ void k_init(unsigned* __restrict__ m_enc, float* __restrict__ den,
                       int n) {
  int i = blockIdx.x * blockDim.x + threadIdx.x;
  if (i < n) { m_enc[i] = ENC_NEG_INF; den[i] = 0.0f; }
}

// ---------------------------------------------------------------------------
// Pass 1: s = LeakyReLU(x @ W + b); store s; atomic segment-max of s.
// One wave = one 16-row tile. A (16x4 f32) per ISA layout:
//   v0: lanes0-15 -> K=0, lanes16-31 -> K=2 ; v1: K=1 / K=3
// B (4x16): W broadcast across all 16 N-columns, so every accumulator lane
// holds the dot result for its row M regardless of its N.
// ---------------------------------------------------------------------------
__global__ __launch_bounds__(256) void k_fused_gemv(
    const float* __restrict__ x, const float* __restrict__ Wv,
    const float* __restrict__ bptr, const int* __restrict__ idx,
    float* __restrict__ s_out, unsigned* __restrict__ m_enc,
    int E, int Bn, int n_tiles)
{
  __shared__ float Wl[F_DIM];
  if (threadIdx.x < F_DIM) Wl[threadIdx.x] = Wv[threadIdx.x];
  __syncthreads();

  const int lane = threadIdx.x & 31;
  const int tile = blockIdx.x * (blockDim.x >> 5) + (threadIdx.x >> 5);
  if (tile >= n_tiles) return;               // wave-uniform: EXEC stays all-1s

  const int hv = lane >> 4;                  // half-wave: 0 -> K{0,1}, 1 -> K{2,3}
  const int l  = lane & 15;
  const long rowBase = (long)tile * 16;

  const float* xrow = x + (rowBase + l) * (long)F_DIM + 2 * hv;
  const float* wrow = Wl + 2 * hv;           // LDS-resident weights
  const float  bias = bptr[0];

  v8f acc = {};
#pragma unroll
  for (int k0 = 0; k0 < F_DIM; k0 += 4) {
    // x streamed exactly once: non-temporal so it doesn't evict L2 state.
    v2f a = __builtin_nontemporal_load((const v2f*)(xrow + k0));
    v2f bm;
    bm.x = wrow[k0];
    bm.y = wrow[k0 + 1];
    acc = __builtin_amdgcn_wmma_f32_16x16x4_f32(
        /*neg_a=*/false, a, /*neg_b=*/false, bm,
        /*c_mod=*/(short)0, acc, /*reuse_a=*/false, /*reuse_b=*/false);
  }

  // C/D layout: lanes0-15 VGPR r -> row M=r ; lanes16-31 VGPR r -> row M=r+8.
  // Lane l==r of each half writes one row -> 16 contiguous stores per tile.
#pragma unroll
  for (int r = 0; r < 8; ++r) {
    if (l == r) {
      float sv = acc[r] + bias;
      sv = (sv >= 0.0f) ? sv : NEG_SLOPE * sv;           // LeakyReLU
      const long row = rowBase + r + 8 * hv;             // flat b*E + e
      s_out[row] = sv;                                   // stays hot in L2
      const int b = (int)(row / E);
      const int e = (int)(row - (long)b * E);
      atomicMax(&m_enc[(long)idx[e] * Bn + b], enc_f32(sv));
    }
  }
}

// ---------------------------------------------------------------------------
// Pass 2: e = exp(s - m[idx]); out = e; atomic segment-sum of e.
// ---------------------------------------------------------------------------
__global__ __launch_bounds__(256) void k_exp(
    const float* __restrict__ s, const int* __restrict__ idx,
    const unsigned* __restrict__ m_enc, float* __restrict__ den,
    float* __restrict__ out, int E, int Bn, int total)
{
  int i = blockIdx.x * blockDim.x + threadIdx.x;
  if (i >= total) return;
  int b = i / E;
  int e = i - b * E;
  long seg = (long)idx[e] * Bn + b;
  float ev = __expf(s[i] - dec_f32(m_enc[seg]));
  out[i] = ev;
  atomicAdd(&den[seg], ev);
}

// ---------------------------------------------------------------------------
// Pass 3: out /= den[idx].
// ---------------------------------------------------------------------------
__global__ __launch_bounds__(256) void k_norm(
    float* __restrict__ out, const int* __restrict__ idx,
    const float* __restrict__ den, int E, int Bn, int total)
{
  int i = blockIdx.x * blockDim.x + threadIdx.x;
  if (i >= total) return;
  int b = i / E;
  int e = i - b * E;
  out[i] = out[i] / den[(long)idx[e] * Bn + b];
}

// ---------------------------------------------------------------------------
extern "C" void kernel_launch(void* const* d_in, const int* in_sizes, int n_in,
                              void* d_out, int out_size, void* d_ws,
                              size_t ws_size, hipStream_t stream)
{
  const float* x    = (const float*)d_in[0];   // [B,E,F]
  const float* W    = (const float*)d_in[1];   // [F,1]
  const float* bias = (const float*)d_in[2];   // [1]
  const int*   idx  = (const int*)  d_in[3];   // [E]
  float* out = (float*)d_out;                  // [B,E,1] flat

  const int E  = in_sizes[3];
  const int Fd = in_sizes[1];                  // 128
  const int Bn = in_sizes[0] / (E * Fd);       // 4
  const int total = Bn * E;                    // 3.2M (multiple of 16)
  const int NB = NSEG * Bn;                    // 400k segment slots

  // Workspace: [m_enc: NB u32][den: NB f32][s: total f32]  (~16 MB)
  unsigned* m_enc = (unsigned*)d_ws;
  float*    den   = (float*)((char*)d_ws + (size_t)NB * 4);
  float*    s_buf = (float*)((char*)d_ws + (size_t)NB * 8);

  k_init<<<(NB + 255) / 256, 256, 0, stream>>>(m_enc, den, NB);

  const int n_tiles = total / 16;              // 200,000 waves
  const int wavesPerBlk = 8;                   // 256 threads = 8 wave32
  const int blocks = (n_tiles + wavesPerBlk - 1) / wavesPerBlk;
  k_fused_gemv<<<blocks, 256, 0, stream>>>(x, W, bias, idx, s_buf, m_enc,
                                           E, Bn, n_tiles);

  k_exp<<<(total + 255) / 256, 256, 0, stream>>>(s_buf, idx, m_enc, den, out,
                                                 E, Bn, total);
  k_norm<<<(total + 255) / 256, 256, 0, stream>>>(out, idx, den, E, Bn, total);
}